// SSIMLoss_15315853377684
// MI455X (gfx1250) — compile-verified
//
#include <hip/hip_runtime.h>

// ---------------------------------------------------------------------------
// SSIM loss, MI455X (gfx1250). One wave32 per 16x16 output tile.
// Separable 7x7 Gaussian blur done as two 16x16xK fp32 WMMA matmuls:
//   tmp = G * X_halo   (vertical,   G[r][j] = g[j-r], 16x22 band)
//   out = tmp * G'     (horizontal, G'[c][x] = g[c-x], 22x16 band)
// K padded to 24 -> 6 chunks of V_WMMA_F32_16X16X4_F32 per pass.
// Band coefficients come from a zero-padded LDS table (one ds_load per
// fragment component) instead of runtime-indexed register arrays, which
// previously lowered to rematerialized v_cndmask select trees.
// ---------------------------------------------------------------------------

typedef __attribute__((ext_vector_type(2))) float v2f;
typedef __attribute__((ext_vector_type(8))) float v8f;

#define IMG_H 256
#define IMG_W 256
#define N_IMG 128
#define TILE  16
#define HALO  22      // TILE + 6 halo
#define FROWS 24      // field rows padded to K multiple of 4
#define FCOLS 32      // field cols padded so the second N-half (cols 16..31) is safe
#define TCOLS 33      // tmp row stride (odd -> spreads LDS banks on column reads)
#define KCH   6       // ceil(22/4) K-chunks of 4
#define GOFF  16      // offset of g[0] inside the padded LDS gauss table
#define GTAB  64      // padded table size (indices 16+k-rM span [1, 39])

__global__ void ssim_init(double* acc) { acc[0] = 0.0; }

__global__ __launch_bounds__(32)
void ssim_tiles(const float* __restrict__ pred,
                const float* __restrict__ target,
                const float* __restrict__ window,
                double* __restrict__ acc)
{
    __shared__ float fld[5][FROWS][FCOLS];   // p, t, p^2, t^2, p*t halos
    __shared__ float tmp[TILE][TCOLS];       // vertical-pass result
    __shared__ float gl[GTAB];               // zero-padded 1-D gaussian table

    const int lane = threadIdx.x;            // 0..31 (wave32)
    const int rM   = lane & 15;              // M (or N) index of this lane
    const int hi   = lane >> 4;              // half-wave select

    const int tilesX      = IMG_W / TILE;            // 16
    const int tilesPerImg = (IMG_H / TILE) * tilesX; // 256
    const int tid = blockIdx.x;
    const int img = tid / tilesPerImg;
    const int t2  = tid % tilesPerImg;
    const int ty  = (t2 / tilesX) * TILE;
    const int tx  = (t2 % tilesX) * TILE;

    const float* __restrict__ p = pred   + (size_t)img * (IMG_H * IMG_W);
    const float* __restrict__ t = target + (size_t)img * (IMG_H * IMG_W);

    // Zero-padded gaussian table. Recover the 1-D gaussian from the
    // (separable) 7x7 window: window[3][i] = g3*g[i], sum_i window[3][i] = g3.
    gl[lane] = 0.f;
    gl[lane + 32] = 0.f;
    {
        float s = 0.f;
        #pragma unroll
        for (int i = 0; i < 7; ++i) s += window[3 * 7 + i];   // uniform -> s_load
        const float inv = 1.0f / s;
        if (lane < 7) gl[GOFF + lane] = window[3 * 7 + lane] * inv;
    }
    __syncthreads();

    // Shared coefficient fragments via LDS table lookup.
    // A (vertical): M=lane%16, K=4*kk+v+2*hi, value g[K-M]
    // B (horizontal): N=lane%16, K=4*kk+v+2*hi, value g[K-N]
    // -> identical per-lane values; out-of-band and K>=HALO both hit the
    //    zero padding (K>=22 implies K-rM >= 7 since rM <= 15).
    v2f Gf[KCH];
    #pragma unroll
    for (int kk = 0; kk < KCH; ++kk) {
        #pragma unroll
        for (int v = 0; v < 2; ++v) {
            const int k = 4 * kk + v + 2 * hi;
            Gf[kk][v] = gl[GOFF + k - rM];
        }
    }

    // Load the 22x22 halo (zero-padded at image borders, matching the
    // reference's zero-pad conv) and build all five fields in LDS.
    for (int idx = lane; idx < FROWS * FCOLS; idx += 32) {
        const int r = idx / FCOLS, c = idx % FCOLS;
        float pv = 0.f, tv = 0.f;
        if (r < HALO && c < HALO) {
            const int y = ty + r - 3, x = tx + c - 3;
            if (y >= 0 && y < IMG_H && x >= 0 && x < IMG_W) {
                pv = p[y * IMG_W + x];
                tv = t[y * IMG_W + x];
            }
        }
        fld[0][r][c] = pv;
        fld[1][r][c] = tv;
        fld[2][r][c] = pv * pv;
        fld[3][r][c] = tv * tv;
        fld[4][r][c] = pv * tv;
    }
    __syncthreads();  // single-wave WG: DS is in-order per wave anyway

    v8f res[5];
    #pragma unroll
    for (int f = 0; f < 5; ++f) {
        // ---- vertical pass: tmp(16 x 32) = G(16 x 24) * fld_f(24 x 32) ----
        #pragma unroll
        for (int nh = 0; nh < 2; ++nh) {
            v8f vacc = {};
            #pragma unroll
            for (int kk = 0; kk < KCH; ++kk) {
                v2f b;
                #pragma unroll
                for (int v = 0; v < 2; ++v)
                    b[v] = fld[f][4 * kk + v + 2 * hi][16 * nh + rM];
                vacc = __builtin_amdgcn_wmma_f32_16x16x4_f32(
                    false, Gf[kk], false, b, (short)0, vacc, false, false);
            }
            // C/D layout: row = v + 8*hi, col = lane%16 (+16 for second N-half)
            #pragma unroll
            for (int v = 0; v < 8; ++v)
                tmp[v + 8 * hi][16 * nh + rM] = vacc[v];
        }
        __syncthreads();

        // ---- horizontal pass: res_f(16 x 16) = tmp(16 x 24) * G'(24 x 16) ----
        v8f hacc = {};
        #pragma unroll
        for (int kk = 0; kk < KCH; ++kk) {
            v2f a;
            #pragma unroll
            for (int v = 0; v < 2; ++v)
                a[v] = tmp[rM][4 * kk + v + 2 * hi];
            hacc = __builtin_amdgcn_wmma_f32_16x16x4_f32(
                false, a, false, Gf[kk], (short)0, hacc, false, false);
        }
        res[f] = hacc;
        __syncthreads();
    }

    // Per-pixel SSIM from the five blurred fields, then tile-local sum.
    const float C1 = 1e-4f;   // 0.01^2
    const float C2 = 9e-4f;   // 0.03^2
    float lsum = 0.f;
    #pragma unroll
    for (int v = 0; v < 8; ++v) {
        const float mu1 = res[0][v], mu2 = res[1][v];
        const float ep2 = res[2][v], et2 = res[3][v], ept = res[4][v];
        const float m11 = mu1 * mu1, m22 = mu2 * mu2, m12 = mu1 * mu2;
        const float s1  = ep2 - m11, s2 = et2 - m22, s12 = ept - m12;
        const float num = (2.f * m12 + C1) * (2.f * s12 + C2);
        const float den = (m11 + m22 + C1) * (s1 + s2 + C2);
        lsum += num / den;
    }
    // wave32 butterfly reduction
    #pragma unroll
    for (int off = 16; off > 0; off >>= 1)
        lsum += __shfl_xor(lsum, off, 32);

    if (lane == 0)
        atomicAdd(acc, (double)lsum);   // f64 accumulate: order noise << f32 ulp

}

__global__ void ssim_fini(const double* __restrict__ acc, float* __restrict__ out)
{
    const double n = (double)N_IMG * (double)IMG_H * (double)IMG_W;
    out[0] = (float)(1.0 - acc[0] / n);
}

extern "C" void kernel_launch(void* const* d_in, const int* in_sizes, int n_in,
                              void* d_out, int out_size, void* d_ws, size_t ws_size,
                              hipStream_t stream)
{
    const float* pred   = (const float*)d_in[0];
    const float* target = (const float*)d_in[1];
    const float* window = (const float*)d_in[2];
    double* acc = (double*)d_ws;
    float*  out = (float*)d_out;

    hipLaunchKernelGGL(ssim_init, dim3(1), dim3(1), 0, stream, acc);

    const int nTiles = N_IMG * (IMG_H / TILE) * (IMG_W / TILE);  // 32768
    hipLaunchKernelGGL(ssim_tiles, dim3(nTiles), dim3(32), 0, stream,
                       pred, target, window, acc);

    hipLaunchKernelGGL(ssim_fini, dim3(1), dim3(1), 0, stream, acc, out);
}